// SAGEIDConv_36000415875687
// MI455X (gfx1250) — compile-verified
//
#include <hip/hip_runtime.h>
#include <hip/hip_bf16.h>

typedef __attribute__((ext_vector_type(2))) float v2f;
typedef __attribute__((ext_vector_type(8))) float v8f;

#define N_NODES 100000
#define N_EDGES 640000
#define D_IN    128
#define D_OUT   128
#define N_ID    10000

// ---------------------------------------------------------------------------
// Kernel 1: zero the scatter workspace (sum [N,128] + count [N]).
// ---------------------------------------------------------------------------
__global__ void zero_ws_kernel(float* __restrict__ ws, int n) {
    int i = blockIdx.x * blockDim.x + threadIdx.x;
    if (i < n) ws[i] = 0.0f;
}

// ---------------------------------------------------------------------------
// Kernel 2: edge scatter-add. One wave32 per edge: 32 lanes x float4 = 128
// floats gathered from node_feature[src], atomically added into sum[dst].
// Lane 0 bumps the degree count. node_feature and sum both fit in L2 (192MB).
// ---------------------------------------------------------------------------
__global__ void scatter_add_kernel(const float* __restrict__ x,
                                   const int* __restrict__ src,
                                   const int* __restrict__ dst,
                                   float* __restrict__ sum,
                                   float* __restrict__ cnt) {
    int edge = (blockIdx.x * blockDim.x + threadIdx.x) >> 5;
    if (edge >= N_EDGES) return;
    int lane = threadIdx.x & 31;
    int s = src[edge];
    int d = dst[edge];
    const float4* xs = reinterpret_cast<const float4*>(x + (size_t)s * D_IN);
    float4 v = xs[lane];
    float* sd = sum + (size_t)d * D_IN + lane * 4;
    atomicAdd(sd + 0, v.x);
    atomicAdd(sd + 1, v.y);
    atomicAdd(sd + 2, v.z);
    atomicAdd(sd + 3, v.w);
    if (lane == 0) atomicAdd(cnt + d, 1.0f);
}

// ---------------------------------------------------------------------------
// Kernel 3: aggr = sum / max(count, 1)  (in place over sum buffer)
// ---------------------------------------------------------------------------
__global__ void normalize_kernel(float* __restrict__ sum,
                                 const float* __restrict__ cnt) {
    int i = blockIdx.x * blockDim.x + threadIdx.x;
    if (i >= N_NODES * D_IN) return;
    float c = cnt[i >> 7];             // i / 128
    float denom = c > 1.0f ? c : 1.0f;
    sum[i] = sum[i] / denom;
}

// ---------------------------------------------------------------------------
// Kernel 4: out[N,128] = [x || aggr] @ W  via v_wmma_f32_16x16x4_f32.
// One wave per 32-row tile (two stacked 16x16x4 A tiles sharing each B
// fragment -> 2 WMMA per B load), full 128-col output: 16 x v8f accumulators.
// K loop split at 128 so the x/aggr pointer select is hoisted out entirely.
// Grid exact (3125 tiles = 625 blocks x 5 waves): EXEC all-1s at every WMMA.
// ---------------------------------------------------------------------------
__global__ void __launch_bounds__(160)
gemm_kernel(const float* __restrict__ x,
            const float* __restrict__ aggr,
            const float* __restrict__ w,
            float* __restrict__ out) {
    const int lane  = threadIdx.x & 31;
    const int wave  = threadIdx.x >> 5;
    const int tile  = blockIdx.x * 5 + wave;        // 0 .. 3124
    const int m0    = tile * 32 + (lane & 15);      // A-tile-0 row
    const int m1    = m0 + 16;                      // A-tile-1 row
    const int kHalf = (lane < 16) ? 0 : 2;          // K sub-offset per half-wave
    const int bN    = lane & 15;                    // B-matrix column

    const float* x0 = x    + (size_t)m0 * D_IN + kHalf;
    const float* x1 = x    + (size_t)m1 * D_IN + kHalf;
    const float* g0 = aggr + (size_t)m0 * D_IN + kHalf;
    const float* g1 = aggr + (size_t)m1 * D_IN + kHalf;

    v8f acc[16];
    #pragma unroll
    for (int i = 0; i < 16; ++i)
        #pragma unroll
        for (int j = 0; j < 8; ++j) acc[i][j] = 0.0f;

    // ---- first half: h[:, k] = x[:, k], W rows [0,128) ----
    for (int k = 0; k < D_IN; k += 4) {
        v2f A0; A0.x = x0[k]; A0.y = x0[k + 1];
        v2f A1; A1.x = x1[k]; A1.y = x1[k + 1];
        const float* wr = w + (size_t)(k + kHalf) * D_OUT + bN;
        #pragma unroll
        for (int nt = 0; nt < 8; ++nt) {
            v2f b;
            b.x = wr[nt * 16];
            b.y = wr[D_OUT + nt * 16];
            acc[nt] = __builtin_amdgcn_wmma_f32_16x16x4_f32(
                false, A0, false, b, (short)0, acc[nt], false, false);
            acc[8 + nt] = __builtin_amdgcn_wmma_f32_16x16x4_f32(
                false, A1, false, b, (short)0, acc[8 + nt], false, false);
        }
    }
    // ---- second half: h[:, 128+k] = aggr[:, k], W rows [128,256) ----
    for (int k = 0; k < D_IN; k += 4) {
        v2f A0; A0.x = g0[k]; A0.y = g0[k + 1];
        v2f A1; A1.x = g1[k]; A1.y = g1[k + 1];
        const float* wr = w + (size_t)(D_IN + k + kHalf) * D_OUT + bN;
        #pragma unroll
        for (int nt = 0; nt < 8; ++nt) {
            v2f b;
            b.x = wr[nt * 16];
            b.y = wr[D_OUT + nt * 16];
            acc[nt] = __builtin_amdgcn_wmma_f32_16x16x4_f32(
                false, A0, false, b, (short)0, acc[nt], false, false);
            acc[8 + nt] = __builtin_amdgcn_wmma_f32_16x16x4_f32(
                false, A1, false, b, (short)0, acc[8 + nt], false, false);
        }
    }

    // C/D layout: VGPR j -> M = j (+8 for upper half-wave), N = lane&15
    const int cRow0 = tile * 32 + ((lane < 16) ? 0 : 8);
    const int cRow1 = cRow0 + 16;
    const int cN    = lane & 15;
    #pragma unroll
    for (int nt = 0; nt < 8; ++nt)
        #pragma unroll
        for (int j = 0; j < 8; ++j) {
            out[(size_t)(cRow0 + j) * D_OUT + nt * 16 + cN] = acc[nt][j];
            out[(size_t)(cRow1 + j) * D_OUT + nt * 16 + cN] = acc[8 + nt][j];
        }
}

// ---------------------------------------------------------------------------
// Kernel 5: id correction. Same WMMA scheme (16-row tile) but A rows are
// gathered through node_id_index and the 16x128 result is scatter-added into
// out with f32 atomics (duplicate ids accumulate, matching .at[].add).
// Grid exact: 625 tiles x 1 wave.
// ---------------------------------------------------------------------------
__global__ void __launch_bounds__(32)
idcorr_kernel(const float* __restrict__ x,
              const float* __restrict__ aggr,
              const float* __restrict__ wid,
              const int* __restrict__ ids,
              float* __restrict__ out) {
    const int lane  = threadIdx.x & 31;
    const int tile  = blockIdx.x;                   // 0 .. 624
    const int r     = ids[tile * 16 + (lane & 15)]; // gathered A row
    const int kHalf = (lane < 16) ? 0 : 2;
    const int bN    = lane & 15;

    const float* xrow = x    + (size_t)r * D_IN + kHalf;
    const float* grow = aggr + (size_t)r * D_IN + kHalf;

    v8f acc[8];
    #pragma unroll
    for (int i = 0; i < 8; ++i)
        #pragma unroll
        for (int j = 0; j < 8; ++j) acc[i][j] = 0.0f;

    for (int k = 0; k < D_IN; k += 4) {
        v2f a; a.x = xrow[k]; a.y = xrow[k + 1];
        const float* wr = wid + (size_t)(k + kHalf) * D_OUT + bN;
        #pragma unroll
        for (int nt = 0; nt < 8; ++nt) {
            v2f b;
            b.x = wr[nt * 16];
            b.y = wr[D_OUT + nt * 16];
            acc[nt] = __builtin_amdgcn_wmma_f32_16x16x4_f32(
                false, a, false, b, (short)0, acc[nt], false, false);
        }
    }
    for (int k = 0; k < D_IN; k += 4) {
        v2f a; a.x = grow[k]; a.y = grow[k + 1];
        const float* wr = wid + (size_t)(D_IN + k + kHalf) * D_OUT + bN;
        #pragma unroll
        for (int nt = 0; nt < 8; ++nt) {
            v2f b;
            b.x = wr[nt * 16];
            b.y = wr[D_OUT + nt * 16];
            acc[nt] = __builtin_amdgcn_wmma_f32_16x16x4_f32(
                false, a, false, b, (short)0, acc[nt], false, false);
        }
    }

    const int cHalf = (lane < 16) ? 0 : 8;
    const int cN    = lane & 15;
    #pragma unroll
    for (int j = 0; j < 8; ++j) {
        const int outRow = ids[tile * 16 + cHalf + j];
        float* orow = out + (size_t)outRow * D_OUT + cN;
        #pragma unroll
        for (int nt = 0; nt < 8; ++nt)
            atomicAdd(orow + nt * 16, acc[nt][j]);
    }
}

// ---------------------------------------------------------------------------
extern "C" void kernel_launch(void* const* d_in, const int* in_sizes, int n_in,
                              void* d_out, int out_size, void* d_ws, size_t ws_size,
                              hipStream_t stream) {
    const float* x   = (const float*)d_in[0];          // [100000,128]
    const int*   ei  = (const int*)d_in[1];            // [2,640000]
    const int*   ids = (const int*)d_in[2];            // [10000]
    const float* w   = (const float*)d_in[3];          // [256,128]
    const float* wid = (const float*)d_in[4];          // [256,128]
    float* out = (float*)d_out;                        // [100000,128]

    float* sum = (float*)d_ws;                         // [100000,128]
    float* cnt = sum + (size_t)N_NODES * D_IN;         // [100000]
    const int* src = ei;
    const int* dst = ei + N_EDGES;

    // 1) zero sum + count (deterministic per call)
    {
        int n = N_NODES * D_IN + N_NODES;
        zero_ws_kernel<<<(n + 255) / 256, 256, 0, stream>>>((float*)d_ws, n);
    }
    // 2) scatter-add: 1 wave per edge, 8 waves per block
    scatter_add_kernel<<<N_EDGES / 8, 256, 0, stream>>>(x, src, dst, sum, cnt);
    // 3) normalize into aggr (in place)
    normalize_kernel<<<(N_NODES * D_IN + 255) / 256, 256, 0, stream>>>(sum, cnt);
    // 4) main GEMM: 3125 x 32-row tiles, 5 waves per block
    gemm_kernel<<<3125 / 5, 160, 0, stream>>>(x, sum, w, out);
    // 5) id-corrected GEMM + atomic scatter: 625 tiles, 1 wave per block
    idcorr_kernel<<<N_ID / 16, 32, 0, stream>>>(x, sum, wid, ids, out);
}